// DiseaseGNN_28578712387810
// MI455X (gfx1250) — compile-verified
//
#include <hip/hip_runtime.h>

// CDNA5 / gfx1250 GCN (2-layer) implementation.
// Layer1: h1pre = x @ W1 via V_WMMA_F32_16X16X4_F32 (fp32 matrix cores),
//         normalized scatter-add over edges (+self loops), +b1, ReLU.
// Layer2: matvec h2 = h1 @ W2, normalized scalar scatter-add, +b2.
//
// Workspace layout (floats), total = 2*N*128 + 2*N = ~103.2 MB for N=100000:
//   agg   [N*128]  layer-1 aggregation accumulator (then holds relu(h1))
//   h1pre [N*128]  x @ W1
//   dinv  [N]      degree -> d^{-1/2} in place
//   h2    [N]      h1 @ W2

typedef float v2f __attribute__((ext_vector_type(2)));
typedef float v8f __attribute__((ext_vector_type(8)));
typedef int   v4i __attribute__((ext_vector_type(4)));

#define F_IN 256
#define F_H  128
#define WLDS_STRIDE 144   // 128 + 16 pad: rows k and k+2 (2*144 dwords apart =>
                          // bank shift 32) land on disjoint bank groups
#define KCHUNK 64         // W1 K-rows staged in LDS per chunk (36 KB LDS)

#ifndef __has_builtin
#define __has_builtin(x) 0
#endif
#if __has_builtin(__builtin_amdgcn_global_load_async_to_lds_b128)
#define HAVE_ASYNC_LDS 1
typedef __attribute__((address_space(1))) v4i gas_v4i;   // global int4*
typedef __attribute__((address_space(3))) v4i las_v4i;   // LDS int4*
#else
#define HAVE_ASYNC_LDS 0
#endif

// ---------------- init: agg=0, deg=1 (self loop), out=b2 -----------------
__global__ void gcn_init(float* __restrict__ agg, float* __restrict__ deg,
                         float* __restrict__ out, const float* __restrict__ b2,
                         long long n128, int n) {
  long long i = (long long)blockIdx.x * 256 + threadIdx.x;
  if (i < n128) agg[i] = 0.f;
  if (i < n) { deg[i] = 1.f; out[i] = b2[0]; }
}

// ---------------- degree: deg[dst] += 1 over real edges ------------------
__global__ void gcn_degree(const long long* __restrict__ eidx,
                           float* __restrict__ deg, long long E) {
  long long e = (long long)blockIdx.x * 256 + threadIdx.x;
  if (e < E) atomicAdd(&deg[eidx[E + e]], 1.f);
}

// ---------------- dinv = rsqrt(deg) in place -----------------------------
__global__ void gcn_rsqrt(float* __restrict__ deg, int n) {
  int i = blockIdx.x * 256 + threadIdx.x;
  if (i < n) { float d = deg[i]; deg[i] = d > 0.f ? rsqrtf(d) : 0.f; }
}

// ---------------- GEMM1: h1pre = x @ W1 (fp32 WMMA) ----------------------
// 8 waves/block; wave w computes rows [m0, m0+16) x all 128 cols.
// W1 staged in LDS in 64-row chunks (async DMA when available);
// 16x16x4 f32 WMMA per (k-step, n-tile).
__global__ __launch_bounds__(256) void gcn_gemm1_wmma(
    const float* __restrict__ x, const float* __restrict__ W1,
    float* __restrict__ h1pre, int nrows) {
  __shared__ float wlds[KCHUNK * WLDS_STRIDE];

  const int wave = threadIdx.x >> 5;
  const int lane = threadIdx.x & 31;
  const int m0   = (blockIdx.x * 8 + wave) * 16;
  const bool active = (m0 < nrows);           // wave-uniform -> EXEC all ones

  const int half = lane >> 4;                 // lane-half: 0 or 1
  const int l16  = lane & 15;
  const int mrow = m0 + l16;                  // A-matrix row for this lane

  v8f acc[8];
#pragma unroll
  for (int t = 0; t < 8; ++t) acc[t] = (v8f){0.f,0.f,0.f,0.f,0.f,0.f,0.f,0.f};

  // lane l of A holds x[mrow][k0 + 2*half + {0,1}]
  const float* xrow = active ? (x + (size_t)mrow * F_IN + 2 * half) : x;

  for (int kc = 0; kc < F_IN; kc += KCHUNK) {
    __syncthreads();
    // stage W1 rows [kc, kc+KCHUNK) into padded LDS
#if HAVE_ASYNC_LDS
    // async DMA: 16B per lane per issue, global row-major -> padded LDS rows
    for (int g4 = threadIdx.x * 4; g4 < KCHUNK * F_H; g4 += 256 * 4) {
      int kk = g4 >> 7;            // /128
      int nn = g4 & (F_H - 1);     // multiple of 4 -> 16B aligned both sides
      __builtin_amdgcn_global_load_async_to_lds_b128(
          (gas_v4i*)(W1 + (size_t)(kc + kk) * F_H + nn),
          (las_v4i*)(&wlds[kk * WLDS_STRIDE + nn]), 0, 0);
    }
#if __has_builtin(__builtin_amdgcn_s_wait_asynccnt)
    __builtin_amdgcn_s_wait_asynccnt(0);
#else
    asm volatile("s_wait_asynccnt 0" ::: "memory");
#endif
#else
    for (int i = threadIdx.x; i < KCHUNK * F_H; i += 256) {
      int kk = i >> 7;
      int nn = i & (F_H - 1);
      wlds[kk * WLDS_STRIDE + nn] = W1[(size_t)(kc + kk) * F_H + nn];
    }
#endif
    __syncthreads();

    if (active) {
      __builtin_prefetch(xrow + kc + KCHUNK, 0, 0);   // global_prefetch_b8
#pragma unroll 4
      for (int k0 = 0; k0 < KCHUNK; k0 += 4) {
        float2 av = *(const float2*)(xrow + kc + k0); // 8B-aligned
        v2f a; a.x = av.x; a.y = av.y;
        const int lk = k0 + 2 * half;                 // B K-row for this half
#pragma unroll
        for (int t = 0; t < 8; ++t) {                 // 8 N-tiles of 16
          const int n = t * 16 + l16;
          v2f b;
          b.x = wlds[lk * WLDS_STRIDE + n];
          b.y = wlds[(lk + 1) * WLDS_STRIDE + n];
          acc[t] = __builtin_amdgcn_wmma_f32_16x16x4_f32(
              false, a, false, b, (short)0, acc[t], false, false);
        }
      }
    }
  }

  if (active) {
    // C/D layout: VGPR r holds M = m0 + r + 8*half, N = l16 (+16*t)
    const int mbase = m0 + 8 * half;
#pragma unroll
    for (int t = 0; t < 8; ++t)
#pragma unroll
      for (int r = 0; r < 8; ++r)
        h1pre[(size_t)(mbase + r) * F_H + t * 16 + l16] = acc[t][r];
  }
}

// ---------------- scatter1: agg[dst] += h1pre[src] * norm ----------------
// One wave32 per edge: float4 per lane (one coalesced 512B row sweep),
// 4 atomic f32 adds per lane. Edges [E, E+N) are the self loops.
__global__ __launch_bounds__(256) void gcn_scatter1(
    const long long* __restrict__ eidx, const float* __restrict__ h1pre,
    const float* __restrict__ dinv, float* __restrict__ agg,
    long long E, long long total) {
  long long e = (long long)blockIdx.x * 8 + (threadIdx.x >> 5);
  if (e >= total) return;
  int lane = threadIdx.x & 31;
  long long s, d;
  if (e < E) { s = eidx[e]; d = eidx[E + e]; } else { s = e - E; d = s; }
  float nrm = dinv[s] * dinv[d];
  float4 v = *(const float4*)(h1pre + s * F_H + lane * 4);
  float* dst = agg + d * F_H + lane * 4;
  atomicAdd(dst + 0, v.x * nrm);
  atomicAdd(dst + 1, v.y * nrm);
  atomicAdd(dst + 2, v.z * nrm);
  atomicAdd(dst + 3, v.w * nrm);
}

// ---------------- bias + relu in place -----------------------------------
__global__ void gcn_bias_relu(float* __restrict__ agg,
                              const float* __restrict__ b1, long long n128) {
  long long i = (long long)blockIdx.x * 256 + threadIdx.x;
  if (i >= n128) return;
  float v = agg[i] + b1[i & (F_H - 1)];
  agg[i] = v > 0.f ? v : 0.f;
}

// ---------------- matvec: h2 = h1 @ W2 (one wave32 per node) -------------
__global__ __launch_bounds__(256) void gcn_matvec(
    const float* __restrict__ h1, const float* __restrict__ W2,
    float* __restrict__ h2, int n) {
  int wave = threadIdx.x >> 5, lane = threadIdx.x & 31;
  int node = blockIdx.x * 8 + wave;
  if (node >= n) return;
  float4 hv = *(const float4*)(h1 + (size_t)node * F_H + lane * 4);
  float4 wv = *(const float4*)(W2 + lane * 4);
  float v = hv.x * wv.x + hv.y * wv.y + hv.z * wv.z + hv.w * wv.w;
#pragma unroll
  for (int off = 16; off > 0; off >>= 1) v += __shfl_xor(v, off, 32);
  if (lane == 0) h2[node] = v;
}

// ---------------- scatter2: out[dst] += h2[src] * norm -------------------
__global__ void gcn_scatter2(const long long* __restrict__ eidx,
                             const float* __restrict__ h2,
                             const float* __restrict__ dinv,
                             float* __restrict__ out,
                             long long E, long long total) {
  long long e = (long long)blockIdx.x * 256 + threadIdx.x;
  if (e >= total) return;
  long long s, d;
  if (e < E) { s = eidx[e]; d = eidx[E + e]; } else { s = e - E; d = s; }
  atomicAdd(&out[d], h2[s] * dinv[s] * dinv[d]);
}

extern "C" void kernel_launch(void* const* d_in, const int* in_sizes, int n_in,
                              void* d_out, int out_size, void* d_ws, size_t ws_size,
                              hipStream_t stream) {
  const float*     x    = (const float*)d_in[0];
  const long long* eidx = (const long long*)d_in[1];   // int64 [2, E] row-major
  const float*     W1   = (const float*)d_in[2];
  const float*     b1   = (const float*)d_in[3];
  const float*     W2   = (const float*)d_in[4];
  const float*     b2   = (const float*)d_in[5];
  float* out = (float*)d_out;

  const int       n     = in_sizes[0] / F_IN;          // 100000
  const long long E     = (long long)in_sizes[1] / 2;  // 3.2M
  const long long total = E + n;                       // edges + self loops
  const long long n128  = (long long)n * F_H;

  float* agg   = (float*)d_ws;          // N*128
  float* h1pre = agg + n128;            // N*128
  float* dinv  = h1pre + n128;          // N (deg -> dinv in place)
  float* h2    = dinv + n;              // N

  gcn_init<<<(unsigned)((n128 + 255) / 256), 256, 0, stream>>>(agg, dinv, out, b2, n128, n);
  gcn_degree<<<(unsigned)((E + 255) / 256), 256, 0, stream>>>(eidx, dinv, E);
  gcn_rsqrt<<<(n + 255) / 256, 256, 0, stream>>>(dinv, n);
  gcn_gemm1_wmma<<<((n + 15) / 16 + 7) / 8, 256, 0, stream>>>(x, W1, h1pre, n);
  gcn_scatter1<<<(unsigned)((total + 7) / 8), 256, 0, stream>>>(eidx, h1pre, dinv, agg, E, total);
  gcn_bias_relu<<<(unsigned)((n128 + 255) / 256), 256, 0, stream>>>(agg, b1, n128);
  gcn_matvec<<<(n + 7) / 8, 256, 0, stream>>>(agg, W2, h2, n);
  gcn_scatter2<<<(unsigned)((total + 255) / 256), 256, 0, stream>>>(eidx, h2, dinv, out, E, total);
}